// AIN_17446157157092
// MI455X (gfx1250) — compile-verified
//
#include <hip/hip_runtime.h>
#include <math.h>

// Problem constants (match reference).
#define NROWS 500000
#define NCOLS 256
#define NSEG  16

#define NB1   512                              // blocks for segment reductions
#define ROWS6 512                              // rows per block in column-sum pass
#define NB2   ((NROWS + ROWS6 - 1) / ROWS6)    // 977

typedef float v2f __attribute__((ext_vector_type(2)));
typedef float v4f __attribute__((ext_vector_type(4)));
typedef float v8f __attribute__((ext_vector_type(8)));

// Order-preserving float<->uint encoding for deterministic max via integer atomics.
__device__ __forceinline__ unsigned f2ord(float f) {
    unsigned u = __float_as_uint(f);
    return (u & 0x80000000u) ? ~u : (u | 0x80000000u);
}
__device__ __forceinline__ float ord2f(unsigned k) {
    unsigned u = (k & 0x80000000u) ? (k & 0x7fffffffu) : ~k;
    return __uint_as_float(u);
}

// ---------------------------------------------------------------------------
// K1: dual GEMV via V_WMMA_F32_16X16X4_F32.
// Each wave handles 16 rows; B has w_local in column 0, w_global in column 1.
// 64 chained WMMAs accumulate the full K=256 dot products in f32.
// ---------------------------------------------------------------------------
__global__ __launch_bounds__(256) void k1_gemv_wmma(
    const float* __restrict__ feats,
    const float* __restrict__ wl, const float* __restrict__ bl,
    const float* __restrict__ wg, const float* __restrict__ bg,
    float* __restrict__ lw, float* __restrict__ gw)
{
    const int lane = threadIdx.x & 31;
    const int wave = threadIdx.x >> 5;
    const int rowBase = blockIdx.x * 128 + wave * 16;     // wave-uniform
    if (rowBase >= NROWS) return;                          // uniform exit: EXEC stays full

    const int m  = lane & 15;          // A row / B,D column for this lane
    const int kh = (lane >> 4) << 1;   // K sub-offset: 0 (lanes 0-15) or 2 (lanes 16-31)

    int row = rowBase + m;
    if (row >= NROWS) row = NROWS - 1; // clamp: garbage only pollutes unstored D rows
    const float* ap = feats + (size_t)row * NCOLS + kh;

    v8f acc = {0.f, 0.f, 0.f, 0.f, 0.f, 0.f, 0.f, 0.f};

    for (int kk = 0; kk < NCOLS; kk += 4) {
        v2f a = *(const v2f*)(ap + kk);                    // A frag: K=kh, kh+1
        float bx = 0.f, by = 0.f;                          // B frag: col m, rows kh/kh+1
        if (m == 0)      { bx = wl[kk + kh]; by = wl[kk + kh + 1]; }
        else if (m == 1) { bx = wg[kk + kh]; by = wg[kk + kh + 1]; }
        v2f b = {bx, by};
        acc = __builtin_amdgcn_wmma_f32_16x16x4_f32(
            false, a, false, b, (short)0, acc, false, false);
    }

    // D layout: acc[j] = row (rowBase + j + 8*(lane>=16)), column = lane&15.
    const int mOff = (lane >> 4) << 3;
    if (m == 0) {
        const float bias = bl[0];
        #pragma unroll
        for (int j = 0; j < 8; ++j) {
            int r = rowBase + mOff + j;
            if (r < NROWS) lw[r] = acc[j] + bias;
        }
    } else if (m == 1) {
        const float bias = bg[0];
        #pragma unroll
        for (int j = 0; j < 8; ++j) {
            int r = rowBase + mOff + j;
            if (r < NROWS) gw[r] = acc[j] + bias;
        }
    }
}

// ---------------------------------------------------------------------------
// K2/K3: segment max (integer-encoded -> order-invariant, deterministic).
// ---------------------------------------------------------------------------
__global__ __launch_bounds__(256) void k2_segmax_part(
    const float* __restrict__ gw, const int* __restrict__ seg,
    unsigned* __restrict__ pmax)
{
    __shared__ unsigned sm[NSEG];
    if (threadIdx.x < NSEG) sm[threadIdx.x] = 0u;
    __syncthreads();
    for (int i = blockIdx.x * 256 + threadIdx.x; i < NROWS; i += NB1 * 256)
        atomicMax(&sm[seg[i]], f2ord(gw[i]));
    __syncthreads();
    if (threadIdx.x < NSEG) pmax[blockIdx.x * NSEG + threadIdx.x] = sm[threadIdx.x];
}

__global__ void k3_segmax_final(const unsigned* __restrict__ pmax,
                                float* __restrict__ segmax)
{
    const int t = threadIdx.x;
    if (t < NSEG) {
        unsigned k = 0u;
        for (int b = 0; b < NB1; ++b) {
            unsigned v = pmax[b * NSEG + t];
            k = (v > k) ? v : k;
        }
        segmax[t] = ord2f(k);
    }
}

// ---------------------------------------------------------------------------
// K4: per-block partial sums of exp(g - segmax) and sigmoid(lw)*exp(...),
// per segment. Fixed-order tree reduction in LDS -> deterministic.
// ---------------------------------------------------------------------------
__global__ __launch_bounds__(256) void k4_segexp_part(
    const float* __restrict__ lw, const float* __restrict__ gw,
    const int* __restrict__ seg, const float* __restrict__ segmax,
    float* __restrict__ psum, float* __restrict__ ptsum)
{
    __shared__ float smax[NSEG];
    __shared__ float red[256 * NSEG];
    if (threadIdx.x < NSEG) smax[threadIdx.x] = segmax[threadIdx.x];
    __syncthreads();

    float accE[NSEG], accT[NSEG];
    #pragma unroll
    for (int s = 0; s < NSEG; ++s) { accE[s] = 0.f; accT[s] = 0.f; }

    for (int i = blockIdx.x * 256 + threadIdx.x; i < NROWS; i += NB1 * 256) {
        const int sv = seg[i];
        const float ex = expf(gw[i] - smax[sv]);
        const float tv = ex / (1.f + expf(-lw[i]));   // sigmoid(lw)*ex
        #pragma unroll
        for (int s = 0; s < NSEG; ++s) {              // select, no scratch indexing
            float msk = (s == sv) ? 1.f : 0.f;
            accE[s] += msk * ex;
            accT[s] += msk * tv;
        }
    }

    // reduce accE
    #pragma unroll
    for (int s = 0; s < NSEG; ++s) red[threadIdx.x * NSEG + s] = accE[s];
    __syncthreads();
    for (int stride = 128; stride > 0; stride >>= 1) {
        if ((int)threadIdx.x < stride)
            #pragma unroll
            for (int s = 0; s < NSEG; ++s)
                red[threadIdx.x * NSEG + s] += red[(threadIdx.x + stride) * NSEG + s];
        __syncthreads();
    }
    if (threadIdx.x < NSEG) psum[blockIdx.x * NSEG + threadIdx.x] = red[threadIdx.x];
    __syncthreads();

    // reduce accT
    #pragma unroll
    for (int s = 0; s < NSEG; ++s) red[threadIdx.x * NSEG + s] = accT[s];
    __syncthreads();
    for (int stride = 128; stride > 0; stride >>= 1) {
        if ((int)threadIdx.x < stride)
            #pragma unroll
            for (int s = 0; s < NSEG; ++s)
                red[threadIdx.x * NSEG + s] += red[(threadIdx.x + stride) * NSEG + s];
        __syncthreads();
    }
    if (threadIdx.x < NSEG) ptsum[blockIdx.x * NSEG + threadIdx.x] = red[threadIdx.x];
}

// K5: segsum, alpha = 1/segsum, W = sum_s T_s/segsum_s  (W = sum of weights).
__global__ void k5_finalize(const float* __restrict__ psum,
                            const float* __restrict__ ptsum,
                            float* __restrict__ alpha, float* __restrict__ Wsum)
{
    __shared__ float sh[NSEG];
    const int t = threadIdx.x;
    if (t < NSEG) {
        float se = 0.f, st = 0.f;
        for (int b = 0; b < NB1; ++b) {
            se += psum[b * NSEG + t];
            st += ptsum[b * NSEG + t];
        }
        alpha[t] = (se > 0.f) ? 1.f / se : 0.f;
        sh[t]    = (se > 0.f) ? st / se : 0.f;
    }
    __syncthreads();
    if (t == 0) {
        float W = 0.f;
        for (int s = 0; s < NSEG; ++s) W += sh[s];
        Wsum[0] = W;
    }
}

// K5b: materialize unnormalized per-row weights (avoids redundant transcendentals
// in the bandwidth-bound column-sum pass).
__global__ __launch_bounds__(256) void k5b_weights(
    const float* __restrict__ lw, const float* __restrict__ gw,
    const int* __restrict__ seg, const float* __restrict__ segmax,
    const float* __restrict__ alpha, float* __restrict__ weight)
{
    const int i = blockIdx.x * 256 + threadIdx.x;
    if (i < NROWS) {
        const int s = seg[i];
        weight[i] = (1.f / (1.f + expf(-lw[i]))) * expf(gw[i] - segmax[s]) * alpha[s];
    }
}

// ---------------------------------------------------------------------------
// K6/K7: weighted column sums S1 = sum w*f, S2 = sum w*f^2, then
// mean = S1/W, inv_std = 1/sqrt(S2/W - mean^2). Fixed-order partials.
// ---------------------------------------------------------------------------
__global__ __launch_bounds__(256) void k6_colsum_part(
    const float* __restrict__ feats, const float* __restrict__ weight,
    float* __restrict__ S1p, float* __restrict__ S2p)
{
    const int t  = threadIdx.x;                      // column
    const int r0 = blockIdx.x * ROWS6;
    const int rend = (r0 + ROWS6 < NROWS) ? r0 + ROWS6 : NROWS;
    float s1 = 0.f, s2 = 0.f;
    for (int r = r0; r < rend; ++r) {
        const float w = weight[r];                   // uniform -> scalar load
        const float f = feats[(size_t)r * NCOLS + t]; // coalesced
        const float wf = w * f;
        s1 += wf;
        s2 += wf * f;
    }
    S1p[blockIdx.x * NCOLS + t] = s1;
    S2p[blockIdx.x * NCOLS + t] = s2;
}

__global__ void k7_colsum_final(const float* __restrict__ S1p,
                                const float* __restrict__ S2p,
                                const float* __restrict__ Wsum,
                                float* __restrict__ mean, float* __restrict__ istd)
{
    const int t = threadIdx.x;                       // 256 threads = 256 columns
    float s1 = 0.f, s2 = 0.f;
    for (int b = 0; b < NB2; ++b) {
        s1 += S1p[b * NCOLS + t];
        s2 += S2p[b * NCOLS + t];
    }
    const float W  = Wsum[0];
    const float m  = s1 / W;
    float var = s2 / W - m * m;
    var = (var > 0.f) ? var : 0.f;
    mean[t] = m;
    istd[t] = 1.f / sqrtf(var);
}

// K8: out = (feats - mean) * inv_std, float4-vectorized (N*C/4 = 32M exactly).
__global__ __launch_bounds__(256) void k8_normalize(
    const float* __restrict__ feats, const float* __restrict__ mean,
    const float* __restrict__ istd, float* __restrict__ out)
{
    const long long e4 = (long long)blockIdx.x * 256 + threadIdx.x;
    const int c4 = ((int)(e4 & 63)) << 2;            // starting column of this float4
    v4f f  = *(const v4f*)(feats + e4 * 4);
    v4f mn = *(const v4f*)(mean + c4);
    v4f is = *(const v4f*)(istd + c4);
    v4f o  = (f - mn) * is;
    *(v4f*)(out + e4 * 4) = o;
}

// ---------------------------------------------------------------------------
extern "C" void kernel_launch(void* const* d_in, const int* in_sizes, int n_in,
                              void* d_out, int out_size, void* d_ws, size_t ws_size,
                              hipStream_t stream) {
    const float* feats = (const float*)d_in[0];
    const int*   seg   = (const int*)d_in[1];
    const float* wl    = (const float*)d_in[2];
    const float* bl    = (const float*)d_in[3];
    const float* wg    = (const float*)d_in[4];
    const float* bg    = (const float*)d_in[5];
    float* out = (float*)d_out;

    // Workspace layout (floats), all chunks 16B-aligned.
    float*    ws     = (float*)d_ws;
    float*    lw     = ws;                               // NROWS
    float*    gw     = lw + NROWS;                       // NROWS
    float*    weight = gw + NROWS;                       // NROWS
    unsigned* pmax   = (unsigned*)(weight + NROWS);      // NB1*NSEG
    float*    psum   = (float*)pmax + NB1 * NSEG;        // NB1*NSEG
    float*    ptsum  = psum + NB1 * NSEG;                // NB1*NSEG
    float*    segmax = ptsum + NB1 * NSEG;               // NSEG
    float*    alpha  = segmax + NSEG;                    // NSEG
    float*    Wsum   = alpha + NSEG;                     // padded to 16
    float*    S1p    = Wsum + 16;                        // NB2*NCOLS
    float*    S2p    = S1p + NB2 * NCOLS;                // NB2*NCOLS
    float*    meanv  = S2p + NB2 * NCOLS;                // NCOLS
    float*    istdv  = meanv + NCOLS;                    // NCOLS

    k1_gemv_wmma<<<(NROWS + 127) / 128, 256, 0, stream>>>(feats, wl, bl, wg, bg, lw, gw);
    k2_segmax_part<<<NB1, 256, 0, stream>>>(gw, seg, pmax);
    k3_segmax_final<<<1, 32, 0, stream>>>(pmax, segmax);
    k4_segexp_part<<<NB1, 256, 0, stream>>>(lw, gw, seg, segmax, psum, ptsum);
    k5_finalize<<<1, 32, 0, stream>>>(psum, ptsum, alpha, Wsum);
    k5b_weights<<<(NROWS + 255) / 256, 256, 0, stream>>>(lw, gw, seg, segmax, alpha, weight);
    k6_colsum_part<<<NB2, 256, 0, stream>>>(feats, weight, S1p, S2p);
    k7_colsum_final<<<1, 256, 0, stream>>>(S1p, S2p, Wsum, meanv, istdv);
    k8_normalize<<<(NROWS * (long long)NCOLS / 4) / 256, 256, 0, stream>>>(feats, meanv, istdv, out);
}